// SSIMLoss_33706903339685
// MI455X (gfx1250) — compile-verified
//
#include <hip/hip_runtime.h>
#include <math.h>

// SSIM loss, (16,3,512,512) f32. Bandwidth-bound (~100MB @ 23.3TB/s ~= 4.3us floor).
// Separable 11-tap Gaussian: horizontal pass on VALU from LDS-staged tiles,
// vertical pass as banded GEMM on V_WMMA_F32_16X16X4_F32 (f32 precision kept),
// fused SSIM map in the WMMA C-layout, deterministic two-stage reduction.

#define IMG_H   512
#define IMG_W   512
#define PLANES  48          // 16 * 3
#define TILE    16
#define HALO    5
#define REG     26          // TILE + 2*HALO
#define KPAD    28          // REG padded to multiple of 4 for WMMA K-chunks
#define TILES_X (IMG_W / TILE)   // 32
#define TILES_PER_PLANE (TILES_X * TILES_X) // 1024
#define NUM_TILES (PLANES * TILES_PER_PLANE) // 49152

typedef float v2f __attribute__((ext_vector_type(2)));
typedef float v8f __attribute__((ext_vector_type(8)));

__global__ __launch_bounds__(32)
void ssim_tile_kernel(const float* __restrict__ pred,
                      const float* __restrict__ targ,
                      float* __restrict__ partial)
{
    __shared__ float sp[REG][KPAD];          // staged pred tile (+halo)
    __shared__ float st[REG][KPAD];          // staged target tile (+halo)
    __shared__ float Hq[5][KPAD][TILE];      // horizontal-pass results: p,t,p2,t2,pt
    __shared__ float gsh[16];                // normalized taps; slots 11..15 == 0

    const int lane  = threadIdx.x;
    const int tile  = blockIdx.x;
    const int plane = tile >> 10;            // / 1024
    const int tyx   = tile & 1023;
    const int ty    = tyx >> 5;              // / 32
    const int tx    = tyx & 31;
    const int row0  = ty * TILE - HALO;      // top of staged region (may be <0)
    const int col0  = tx * TILE - HALO;
    const float* __restrict__ pbase = pred + (size_t)plane * (IMG_H * IMG_W);
    const float* __restrict__ tbase = targ + (size_t)plane * (IMG_H * IMG_W);

    // ---- normalized 1D gaussian (sigma = 1.5, window 11), zero-padded to 16 ----
    if (lane < 16) {
        float s = 0.f;
#pragma unroll
        for (int j = 0; j < 11; ++j) {
            float d = (float)(j - 5);
            s += expf(-d * d / 4.5f);
        }
        float v = 0.f;
        if (lane < 11) {
            float d = (float)(lane - 5);
            v = expf(-d * d / 4.5f);
        }
        gsh[lane] = v / s;                   // lanes 11..15 write exact 0
    }
    __syncthreads();

    float g[11];
#pragma unroll
    for (int j = 0; j < 11; ++j) g[j] = gsh[j];

    // ---- stage pred/target region (26x26, zero-padded at image border) ----
    for (int r = 0; r < REG; ++r) {
        if (lane < REG) {
            int gr = row0 + r;
            int gc = col0 + lane;
            bool inb = (gr >= 0) & (gr < IMG_H) & (gc >= 0) & (gc < IMG_W);
            int off = gr * IMG_W + gc;
            sp[r][lane] = inb ? pbase[off] : 0.f;
            st[r][lane] = inb ? tbase[off] : 0.f;
        }
    }
    __syncthreads();

    // ---- horizontal pass (VALU): 26 rows x 16 cols x 5 quantities ----
    for (int i = lane; i < REG * TILE; i += 32) {   // 416 = 13 * 32, uniform
        int r = i >> 4;
        int c = i & 15;
        float hp = 0.f, ht = 0.f, hpp = 0.f, htt = 0.f, hpt = 0.f;
#pragma unroll
        for (int j = 0; j < 11; ++j) {
            float w  = g[j];
            float pv = sp[r][c + j];
            float tv = st[r][c + j];
            hp  += w * pv;
            ht  += w * tv;
            hpp += w * pv * pv;
            htt += w * tv * tv;
            hpt += w * pv * tv;
        }
        Hq[0][r][c] = hp;
        Hq[1][r][c] = ht;
        Hq[2][r][c] = hpp;
        Hq[3][r][c] = htt;
        Hq[4][r][c] = hpt;
    }
    // zero K-pad rows 26..27 so the 7th WMMA chunk contributes nothing
    for (int i = lane; i < 5 * 2 * TILE; i += 32) { // 160 = 5 * 32, uniform
        int q  = i >> 5;
        int rc = i & 31;
        Hq[q][REG + (rc >> 4)][rc & 15] = 0.f;
    }
    __syncthreads();

    // ---- vertical pass as banded GEMM: Y(16x16) = Gv(16x28) * H(28x16) ----
    // A layout (16x4 f32): lane<16 -> row=lane, K=k,k+1 ; lane>=16 -> row=lane-16, K=k+2,k+3
    // B layout (4x16 f32): lane<16 -> col=lane, rows k,k+1 ; lane>=16 -> col=lane-16, rows k+2,k+3
    const int mrow = lane & 15;
    const int koff = (lane >> 4) << 1;   // 0 or 2

    // Hoist all A-matrix values: index clamped into the zero-padded table,
    // so each is a single unconditional ds_load (no EXEC churn near WMMA).
    float av[14];
#pragma unroll
    for (int kc = 0; kc < 7; ++kc) {
        unsigned u0 = (unsigned)(kc * 4 + koff - mrow);
        unsigned u1 = u0 + 1u;
        av[2 * kc]     = gsh[u0 <= 10u ? u0 : 11u];   // Gv[row][kk0]
        av[2 * kc + 1] = gsh[u1 <= 10u ? u1 : 11u];   // Gv[row][kk1]
    }

    v8f zero = {0.f, 0.f, 0.f, 0.f, 0.f, 0.f, 0.f, 0.f};
    v8f accP = zero, accT = zero, accPP = zero, accTT = zero, accPT = zero;

#pragma unroll
    for (int kc = 0; kc < 7; ++kc) {
        int kk0 = kc * 4 + koff;
        int kk1 = kk0 + 1;
        v2f a;
        a.x = av[2 * kc];
        a.y = av[2 * kc + 1];

#define DO_WMMA(ACC, Q)                                                        \
        {                                                                      \
            v2f b;                                                             \
            b.x = Hq[Q][kk0][mrow];                                            \
            b.y = Hq[Q][kk1][mrow];                                            \
            ACC = __builtin_amdgcn_wmma_f32_16x16x4_f32(                       \
                false, a, false, b, (short)0, ACC, false, false);              \
        }
        DO_WMMA(accP, 0)
        DO_WMMA(accT, 1)
        DO_WMMA(accPP, 2)
        DO_WMMA(accTT, 3)
        DO_WMMA(accPT, 4)
#undef DO_WMMA
    }

    // ---- fused SSIM map in C-layout (8 elements per lane) + wave reduce ----
    const float C1v = 1.0e-4f;   // 0.01^2
    const float C2v = 9.0e-4f;   // 0.03^2
    float s = 0.f;
#pragma unroll
    for (int e = 0; e < 8; ++e) {
        float mp  = accP[e];
        float mt  = accT[e];
        float mp2 = mp * mp;
        float mt2 = mt * mt;
        float mpt = mp * mt;
        float sp2 = accPP[e] - mp2;
        float st2 = accTT[e] - mt2;
        float spt = accPT[e] - mpt;
        float num = (2.f * mpt + C1v) * (2.f * spt + C2v);
        float den = (mp2 + mt2 + C1v) * (sp2 + st2 + C2v);
        s += num / den;
    }
#pragma unroll
    for (int off = 16; off > 0; off >>= 1)
        s += __shfl_xor(s, off, 32);
    if (lane == 0)
        partial[tile] = s;
}

// Deterministic fixed-order final reduction: 1 - mean(ssim_map)
__global__ __launch_bounds__(256)
void ssim_reduce_kernel(const float* __restrict__ partial,
                        float* __restrict__ out, int n, float invN)
{
    __shared__ float red[256];
    int tid = threadIdx.x;
    float s = 0.f;
    for (int i = tid; i < n; i += 256)
        s += partial[i];
    red[tid] = s;
    __syncthreads();
    for (int off = 128; off > 0; off >>= 1) {
        if (tid < off) red[tid] += red[tid + off];
        __syncthreads();
    }
    if (tid == 0)
        out[0] = 1.0f - red[0] * invN;
}

extern "C" void kernel_launch(void* const* d_in, const int* in_sizes, int n_in,
                              void* d_out, int out_size, void* d_ws, size_t ws_size,
                              hipStream_t stream)
{
    const float* pred = (const float*)d_in[0];
    const float* targ = (const float*)d_in[1];
    float* out = (float*)d_out;
    float* partial = (float*)d_ws;   // NUM_TILES floats (192 KiB)

    ssim_tile_kernel<<<NUM_TILES, 32, 0, stream>>>(pred, targ, partial);

    const float invN = 1.0f / (float)(PLANES * IMG_H * IMG_W);
    ssim_reduce_kernel<<<1, 256, 0, stream>>>(partial, out, NUM_TILES, invN);
}